// AttentionFlow_32753420599373
// MI455X (gfx1250) — compile-verified
//
#include <hip/hip_runtime.h>
#include <math.h>

#define DIM   128
#define DS    32
#define BQ    64
#define NSEGC 50000
#define NTGTC 60000

typedef float v2f __attribute__((ext_vector_type(2)));
typedef float v8f __attribute__((ext_vector_type(8)));

#ifndef __has_builtin
#define __has_builtin(x) 0
#endif
#if __has_builtin(__builtin_amdgcn_global_load_async_to_lds_b128)
#define HAVE_ASYNC_LDS 1
#else
#define HAVE_ASYNC_LDS 0
#endif

#define CBAR() asm volatile("" ::: "memory")

#if HAVE_ASYNC_LDS
typedef int v4i __attribute__((vector_size(16)));
typedef __attribute__((address_space(1))) v4i* g4p;
typedef __attribute__((address_space(3))) v4i* l4p;
__device__ __forceinline__ void async_cp16(const void* g, void* l) {
  // 16B per lane, global -> LDS, tracked by ASYNCcnt
  __builtin_amdgcn_global_load_async_to_lds_b128((g4p)g, (l4p)l, 0, 0);
}
#define WAIT_ASYNC() asm volatile("s_wait_asynccnt 0x0" ::: "memory")
#define WAIT_DS()    asm volatile("s_wait_dscnt 0x0" ::: "memory")
#else
#define WAIT_ASYNC() CBAR()
#define WAIT_DS()    CBAR()
#endif

__device__ __forceinline__ v8f wmma4(v2f a, v2f b, v8f c) {
  // D = A(16x4 f32) * B(4x16 f32) + C(16x16 f32)
  return __builtin_amdgcn_wmma_f32_16x16x4_f32(false, a, false, b, (short)0, c, false, false);
}

// monotonic float -> uint key so atomicMax(uint) implements float max
__device__ __forceinline__ unsigned fkey(float f) {
  unsigned u = __float_as_uint(f);
  return (u & 0x80000000u) ? ~u : (u | 0x80000000u);
}
__device__ __forceinline__ float funkey(unsigned k) {
  unsigned u = (k & 0x80000000u) ? (k & 0x7FFFFFFFu) : ~k;
  return __uint_as_float(u);
}

// ---------------- init: zero reduction buffers + output ----------------
__global__ void af_init_kernel(unsigned* __restrict__ segmax, float* __restrict__ segsum,
                               float* __restrict__ egsum, float* __restrict__ out) {
  int i = blockIdx.x * blockDim.x + threadIdx.x;
  if (i < NSEGC) { segmax[i] = 0u; segsum[i] = 0.f; }
  if (i < NTGTC) out[i] = 0.f;
  if (i < BQ)    egsum[i] = 0.f;
}

// ---------------- query precompute, stage 1: qproj[3][64][32] ----------------
__global__ void af_qproj_kernel(const float* __restrict__ qsrc, const float* __restrict__ qrel,
                                const float* __restrict__ qtime,
                                const float* __restrict__ proj_w, const float* __restrict__ proj_b,
                                float* __restrict__ qproj) {
  int t = blockIdx.x * blockDim.x + threadIdx.x;
  if (t >= 3 * BQ * DS) return;
  int i = t & (DS - 1);
  int b = (t >> 5) & (BQ - 1);
  int p = t / (BQ * DS);
  const float* row = (p == 0 ? qsrc : (p == 1 ? qrel : qtime)) + (size_t)b * DIM;
  float s = proj_b[i];
  for (int d = 0; d < DIM; ++d) s += row[d] * proj_w[d * DS + i];
  qproj[t] = s;
}

// ---------------- query precompute, stage 2: qL/qR [64][32] ----------------
__global__ void af_qlr_kernel(const float* __restrict__ qproj,
                              const float* __restrict__ left_w, const float* __restrict__ left_b,
                              const float* __restrict__ right_w, const float* __restrict__ right_b,
                              float* __restrict__ qL, float* __restrict__ qR) {
  int t = blockIdx.x * blockDim.x + threadIdx.x;
  if (t >= 2 * BQ * DS) return;
  int j = t & (DS - 1);
  int b = (t >> 5) & (BQ - 1);
  int side = t / (BQ * DS);
  const float* W = side ? right_w : left_w;
  const float* bias = side ? right_b : left_b;
  float s = bias[j];
  for (int tt = 0; tt < 3 * DS; ++tt)
    s += qproj[(tt / DS) * BQ * DS + b * DS + (tt % DS)] * W[(2 * DS + tt) * DS + j];
  (side ? qR : qL)[b * DS + j] = s;
}

// ---------------- fused edge kernel: async-stage + project + MLP + logit + seg-max ----------------
// 128 threads = 4 waves; each wave owns a 16-edge tile (64 edges / block).
__global__ __launch_bounds__(128) void af_edge_kernel(
    const float* __restrict__ natt,
    const float* __restrict__ hvi, const float* __restrict__ hvj, const float* __restrict__ rel,
    const float* __restrict__ proj_w, const float* __restrict__ proj_b,
    const float* __restrict__ left_w, const float* __restrict__ right_w,
    const float* __restrict__ center_w, const float* __restrict__ center_b,
    const int* __restrict__ eg_idx, const int* __restrict__ seg_vi,
    const float* __restrict__ qLg, const float* __restrict__ qRg,
    float* __restrict__ ws_att, unsigned* __restrict__ segmax, int E) {
  extern __shared__ float smem[];
  float* PW = smem;               // 128*32
  float* LW = PW + 4096;          // left_w rows 0..63
  float* RW = LW + 2048;          // right_w rows 0..63
  float* CW = RW + 2048;          // center_w
  float* CB = CW + 1024;          // center_b
  float* QL = CB + 32;            // 64*32
  float* QR = QL + 2048;          // 64*32
  int*   EG = (int*)(QR + 2048);  // 4 waves * 16
  float* ST = (float*)(EG + 64);  // stA: 4 * 16*33
  float* SU = ST + 4 * 528;       // stB: 4 * 16*33
  float* RA = SU + 4 * 528;       // raw tile A: 4 * 16*132
  float* RB = RA + 4 * 2112;      // raw tile B: 4 * 16*132

  const int tid  = threadIdx.x;
  const int wave = tid >> 5;
  const int lane = tid & 31;
  const int m  = lane & 15;   // A-layout row / D-layout column index
  const int h  = lane >> 4;
  const int n0 = m;
  const int base = blockIdx.x * 64 + wave * 16;

  float* stA  = ST + wave * 528;
  float* stB  = SU + wave * 528;
  float* rawA = RA + wave * 2112;
  float* rawB = RB + wave * 2112;

  // stream one [16,128] fp32 tile into LDS (row stride 132 floats: DS-read bank-conflict free)
  auto load_tile = [&](const float* __restrict__ X, float* raw) {
#pragma unroll 4
    for (int i = 0; i < 16; ++i) {
      int rr = base + i; if (rr >= E) rr = E - 1;
      const float* g = X + (size_t)rr * DIM + lane * 4;
      float* l = raw + i * 132 + lane * 4;
#if HAVE_ASYNC_LDS
      async_cp16((const void*)g, (void*)l);
#else
      float4 v = *(const float4*)g;
      *(float4*)l = v;
#endif
    }
  };

  // kick off async streams for rel + hvi, then stage weights underneath them
  load_tile(rel, rawB);
  load_tile(hvi, rawA);

  for (int i = tid; i < 4096; i += 128) PW[i] = proj_w[i];
  for (int i = tid; i < 2048; i += 128) { LW[i] = left_w[i]; RW[i] = right_w[i]; }
  for (int i = tid; i < 1024; i += 128) CW[i] = center_w[i];
  if (tid < 32) CB[tid] = center_b[tid];
  for (int i = tid; i < 2048; i += 128) { QL[i] = qLg[i]; QR[i] = qRg[i]; }
  if (lane < 16) {
    int e = base + lane; if (e >= E) e = E - 1;
    EG[wave * 16 + lane] = eg_idx[e];
  }
  __syncthreads();

  const float pb0 = proj_b[n0], pb1 = proj_b[n0 + 16];

  // project [16,128] raw LDS tile -> [16,32] into LDS staging (A-layout friendly)
  auto project_to = [&](const float* raw, float* st) {
    v8f a0, a1;
#pragma unroll
    for (int r = 0; r < 8; ++r) { a0[r] = pb0; a1[r] = pb1; }
#pragma unroll 8
    for (int k = 0; k < 32; ++k) {
      const int kk = 4 * k + 2 * h;
      v2f a;  a.x  = raw[m * 132 + kk];   a.y  = raw[m * 132 + kk + 1];
      v2f b0; b0.x = PW[kk * DS + n0];    b0.y = PW[(kk + 1) * DS + n0];
      a0 = wmma4(a, b0, a0);
      v2f b1; b1.x = PW[kk * DS + n0+16]; b1.y = PW[(kk + 1) * DS + n0+16];
      a1 = wmma4(a, b1, a1);
    }
#pragma unroll
    for (int r = 0; r < 8; ++r) {
      st[(r + 8 * h) * 33 + n0]      = a0[r];
      st[(r + 8 * h) * 33 + n0 + 16] = a1[r];
    }
  };

  // [16,32](LDS A-layout) @ [32,32](LDS) accumulated into (c0,c1)
  auto mm32 = [&](const float* st, const float* W, v8f& c0, v8f& c1) {
#pragma unroll
    for (int k = 0; k < 8; ++k) {
      const int kk = 4 * k + 2 * h;
      v2f a;  a.x  = st[m * 33 + kk];    a.y  = st[m * 33 + kk + 1];
      v2f b0; b0.x = W[kk * DS + n0];    b0.y = W[(kk + 1) * DS + n0];
      c0 = wmma4(a, b0, c0);
      v2f b1; b1.x = W[kk * DS + n0+16]; b1.y = W[(kk + 1) * DS + n0+16];
      c1 = wmma4(a, b1, c1);
    }
  };

  // ---- stage 1: rel_p -> stB, hvi_p -> stA (raw tiles must have landed)
  WAIT_ASYNC();
  project_to(rawB, stB);
  project_to(rawA, stA);
  CBAR();
  // all DS reads of rawA retired -> safe to overwrite it with the hvj stream
  WAIT_DS();
  load_tile(hvj, rawA);   // overlaps with lh compute below
  CBAR();

  // ---- lh = leaky( hvi_p@Lw1 + rel_p@Lw2 + qL[eg] )
  v8f l0, l1;
#pragma unroll
  for (int r = 0; r < 8; ++r) {
    int eg = EG[wave * 16 + r + 8 * h];
    l0[r] = QL[eg * DS + n0];
    l1[r] = QL[eg * DS + n0 + 16];
  }
  mm32(stA, LW, l0, l1);
  mm32(stB, LW + 1024, l0, l1);
#pragma unroll
  for (int r = 0; r < 8; ++r) {
    l0[r] = fmaxf(l0[r], 0.01f * l0[r]);
    l1[r] = fmaxf(l1[r], 0.01f * l1[r]);
  }

  // ---- hvj_p -> stA (hvi_p consumed), rh = leaky( hvj_p@Rw1 + rel_p@Rw2 + qR[eg] )
  WAIT_ASYNC();
  project_to(rawA, stA);
  CBAR();
  v8f r0, r1;
#pragma unroll
  for (int r = 0; r < 8; ++r) {
    int eg = EG[wave * 16 + r + 8 * h];
    r0[r] = QR[eg * DS + n0];
    r1[r] = QR[eg * DS + n0 + 16];
  }
  mm32(stA, RW, r0, r1);
  mm32(stB, RW + 1024, r0, r1);
#pragma unroll
  for (int r = 0; r < 8; ++r) {
    r0[r] = fmaxf(r0[r], 0.01f * r0[r]);
    r1[r] = fmaxf(r1[r], 0.01f * r1[r]);
  }

  // ---- c = rh @ center_w + center_b (restage rh into A layout; rel_p dead -> reuse stB)
  CBAR();
#pragma unroll
  for (int r = 0; r < 8; ++r) {
    stB[(r + 8 * h) * 33 + n0]      = r0[r];
    stB[(r + 8 * h) * 33 + n0 + 16] = r1[r];
  }
  CBAR();
  v8f c0, c1;
#pragma unroll
  for (int r = 0; r < 8; ++r) { c0[r] = CB[n0]; c1[r] = CB[n0 + 16]; }
  mm32(stB, CW, c0, c1);

  // ---- logit = sum_n lh*c ; reduce the 16 D-layout columns via LDS
  CBAR();
#pragma unroll
  for (int r = 0; r < 8; ++r)
    stA[(r + 8 * h) * 33 + n0] = l0[r] * c0[r] + l1[r] * c1[r];
  CBAR();
  if (lane < 16) {
    int ee = base + lane;
    if (ee < E) {
      float s = 0.f;
#pragma unroll
      for (int j = 0; j < 16; ++j) s += stA[lane * 33 + j];
      float att = s * natt[ee];
      ws_att[ee] = att;
      atomicMax(&segmax[seg_vi[ee]], fkey(att));
    }
  }
}

// ---------------- softmax tail passes ----------------
__global__ void af_exp_segsum_kernel(const int* __restrict__ seg_vi,
                                     const unsigned* __restrict__ segmax,
                                     float* __restrict__ ws_att,
                                     float* __restrict__ segsum, int E) {
  int i = blockIdx.x * blockDim.x + threadIdx.x;
  if (i >= E) return;
  int s = seg_vi[i];
  float ex = __expf(ws_att[i] - funkey(segmax[s]));
  ws_att[i] = ex;
  atomicAdd(&segsum[s], ex);
}

__global__ void af_norm_egsum_kernel(const int* __restrict__ seg_vi,
                                     const int* __restrict__ eg_idx,
                                     const float* __restrict__ segsum,
                                     float* __restrict__ ws_att,
                                     float* __restrict__ egsum, int E) {
  __shared__ float bins[BQ];
  if (threadIdx.x < BQ) bins[threadIdx.x] = 0.f;
  __syncthreads();
  int i = blockIdx.x * blockDim.x + threadIdx.x;
  if (i < E) {
    float soft = ws_att[i] / segsum[seg_vi[i]];
    ws_att[i] = soft;
    atomicAdd(&bins[eg_idx[i]], soft);
  }
  __syncthreads();
  if (threadIdx.x < BQ) {
    float v = bins[threadIdx.x];
    if (v != 0.f) atomicAdd(&egsum[threadIdx.x], v);
  }
}

__global__ void af_scatter_kernel(const int* __restrict__ eg_idx,
                                  const int* __restrict__ seg_vj,
                                  const float* __restrict__ egsum,
                                  const float* __restrict__ ws_att,
                                  float* __restrict__ out, int E) {
  int i = blockIdx.x * blockDim.x + threadIdx.x;
  if (i >= E) return;
  float v = ws_att[i] / egsum[eg_idx[i]];
  atomicAdd(&out[seg_vj[i]], v);
}

extern "C" void kernel_launch(void* const* d_in, const int* in_sizes, int n_in,
                              void* d_out, int out_size, void* d_ws, size_t ws_size,
                              hipStream_t stream) {
  const float* natt   = (const float*)d_in[0];
  const float* hvi    = (const float*)d_in[1];
  const float* hvj    = (const float*)d_in[2];
  const float* rel    = (const float*)d_in[3];
  const float* qsrc   = (const float*)d_in[4];
  const float* qrel   = (const float*)d_in[5];
  const float* qtime  = (const float*)d_in[6];
  const float* proj_w = (const float*)d_in[7];
  const float* proj_b = (const float*)d_in[8];
  const float* left_w = (const float*)d_in[9];
  const float* left_b = (const float*)d_in[10];
  const float* right_w= (const float*)d_in[11];
  const float* right_b= (const float*)d_in[12];
  const float* center_w=(const float*)d_in[13];
  const float* center_b=(const float*)d_in[14];
  const int* eg_idx   = (const int*)d_in[15];
  const int* seg_vi   = (const int*)d_in[16];
  const int* seg_vj   = (const int*)d_in[17];
  float* out = (float*)d_out;
  const int E = in_sizes[0];

  // workspace layout
  float*    ws_att = (float*)d_ws;
  unsigned* segmax = (unsigned*)(ws_att + E);
  float*    segsum = (float*)(segmax + NSEGC);
  float*    egsum  = segsum + NSEGC;
  float*    qL     = egsum + BQ;
  float*    qR     = qL + BQ * DS;
  float*    qproj  = qR + BQ * DS;   // 3*64*32

  int ninit = (NSEGC > NTGTC) ? NSEGC : NTGTC;
  af_init_kernel<<<(ninit + 255) / 256, 256, 0, stream>>>(segmax, segsum, egsum, out);

  af_qproj_kernel<<<(3 * BQ * DS + 255) / 256, 256, 0, stream>>>(
      qsrc, qrel, qtime, proj_w, proj_b, qproj);
  af_qlr_kernel<<<(2 * BQ * DS + 255) / 256, 256, 0, stream>>>(
      qproj, left_w, left_b, right_w, right_b, qL, qR);

  size_t shbytes =
      (4096 + 2048 + 2048 + 1024 + 32 + 2048 + 2048 + 64 + 2 * 4 * 528 + 2 * 4 * 2112) *
      sizeof(float);
  af_edge_kernel<<<(E + 63) / 64, 128, shbytes, stream>>>(
      natt, hvi, hvj, rel, proj_w, proj_b, left_w, right_w, center_w, center_b,
      eg_idx, seg_vi, qL, qR, ws_att, segmax, E);

  int gb = (E + 255) / 256;
  af_exp_segsum_kernel<<<gb, 256, 0, stream>>>(seg_vi, segmax, ws_att, segsum, E);
  af_norm_egsum_kernel<<<gb, 256, 0, stream>>>(seg_vi, eg_idx, segsum, ws_att, egsum, E);
  af_scatter_kernel<<<gb, 256, 0, stream>>>(eg_idx, seg_vj, egsum, ws_att, out, E);
}